// GatedRGCN_24567212933530
// MI455X (gfx1250) — compile-verified
//
#include <hip/hip_runtime.h>
#include <hip/hip_bf16.h>

// Gated RGCN for gfx1250 (MI455X), fp32 WMMA path (V_WMMA_F32_16X16X4_F32).
// C = 256 features, R = 4 relations.

typedef __attribute__((ext_vector_type(2))) float v2f;
typedef __attribute__((ext_vector_type(8))) float v8f;

#define FEAT 256

// ---------------------------------------------------------------------------
// Kernel 0: zero a float buffer (cnt), since harness poisons d_ws with 0xAA.
// ---------------------------------------------------------------------------
__global__ __launch_bounds__(256) void zero_f32(float* __restrict__ p, int n) {
    int i = blockIdx.x * blockDim.x + threadIdx.x;
    if (i < n) p[i] = 0.0f;
}

// ---------------------------------------------------------------------------
// Kernel 1: per-(dst, relation) in-degree counts. cnt[d*4 + r] += 1.
// ---------------------------------------------------------------------------
__global__ __launch_bounds__(256) void count_edges(
    const int* __restrict__ dstI, const int* __restrict__ etype,
    float* __restrict__ cnt, int E) {
    int e = blockIdx.x * blockDim.x + threadIdx.x;
    if (e >= E) return;
    int d = dstI[e];
    int r = etype[e];
    atomicAdd(&cnt[d * 4 + r], 1.0f);
}

// ---------------------------------------------------------------------------
// Kernel 2: fused 5-way GEMM via fp32 WMMA.
//   blockIdx.y == 0 : u0    = X @ rootW + bias
//   blockIdx.y == m : trans[m-1] = X @ relW[m-1]
// Block = 128 threads (4 waves). Each wave: 16 rows x 64 cols
// (four 16x16 accumulators at col offsets +0,+64,+128,+192).
// A-frag (16x4 f32): lane<16 row=lane K={k0,k0+1}; lane>=16 K={k0+2,k0+3}.
// B-frag (4x16 f32): VGPR0 = K-row {0|2} striped over cols, VGPR1 = K-row {1|3}.
// C/D (16x16 f32):   acc[v], lanes 0-15 -> row v, lanes 16-31 -> row v+8.
// ---------------------------------------------------------------------------
__global__ __launch_bounds__(128) void rgcn_gemm_wmma(
    const float* __restrict__ X,      // [N, 256]
    const float* __restrict__ rootW,  // [256, 256]
    const float* __restrict__ relW,   // [4, 256, 256]
    const float* __restrict__ bias,   // [256]
    float* __restrict__ u0,           // [N, 256]
    float* __restrict__ trans,        // [4, N, 256]
    int N) {
    const int mat  = blockIdx.y;           // 0=root, 1..4=relation mat-1
    const int rowB = blockIdx.x * 16;
    const int wave = threadIdx.x >> 5;     // 0..3
    const int lane = threadIdx.x & 31;
    const int half = lane >> 4;            // 0 or 1
    const int l16  = lane & 15;

    const float* __restrict__ W =
        (mat == 0) ? rootW : (relW + (size_t)(mat - 1) * FEAT * FEAT);
    float* __restrict__ Out =
        (mat == 0) ? u0 : (trans + (size_t)(mat - 1) * N * FEAT);

    int row = rowB + l16;
    if (row >= N) row = N - 1;             // clamp (N=50000 is a multiple of 16)
    const float* __restrict__ arow = X + (size_t)row * FEAT;

    const int c0 = wave * 16 + l16;        // this lane's column within tile 0

    v8f acc0 = {}, acc1 = {}, acc2 = {}, acc3 = {};

    for (int k0 = 0; k0 < FEAT; k0 += 4) {
        const int ka = k0 + 2 * half;
        v2f a = *(const v2f*)(arow + ka);              // contiguous K pair
        const float* __restrict__ wr0 = W + (size_t)ka * FEAT;
        const float* __restrict__ wr1 = wr0 + FEAT;
        v2f b;
        b.x = wr0[c0];       b.y = wr1[c0];
        acc0 = __builtin_amdgcn_wmma_f32_16x16x4_f32(false, a, false, b,
                                                     (short)0, acc0, false, false);
        b.x = wr0[c0 + 64];  b.y = wr1[c0 + 64];
        acc1 = __builtin_amdgcn_wmma_f32_16x16x4_f32(false, a, false, b,
                                                     (short)0, acc1, false, false);
        b.x = wr0[c0 + 128]; b.y = wr1[c0 + 128];
        acc2 = __builtin_amdgcn_wmma_f32_16x16x4_f32(false, a, false, b,
                                                     (short)0, acc2, false, false);
        b.x = wr0[c0 + 192]; b.y = wr1[c0 + 192];
        acc3 = __builtin_amdgcn_wmma_f32_16x16x4_f32(false, a, false, b,
                                                     (short)0, acc3, false, false);
    }

    float b0 = 0.0f, b1 = 0.0f, b2 = 0.0f, b3 = 0.0f;
    if (mat == 0) {
        b0 = bias[c0];       b1 = bias[c0 + 64];
        b2 = bias[c0 + 128]; b3 = bias[c0 + 192];
    }

#pragma unroll
    for (int v = 0; v < 8; ++v) {
        int r = rowB + v + 8 * half;
        if (r < N) {
            float* __restrict__ orow = Out + (size_t)r * FEAT;
            orow[c0]        = acc0[v] + b0;
            orow[c0 + 64]   = acc1[v] + b1;
            orow[c0 + 128]  = acc2[v] + b2;
            orow[c0 + 192]  = acc3[v] + b3;
        }
    }
}

// ---------------------------------------------------------------------------
// Kernel 3: edge scatter with fused per-relation mean.
// u0[dst] += trans[r][src] / max(cnt[dst][r], 1).
// 64 threads per edge, float4 per thread, 4 global f32 atomics each.
// ---------------------------------------------------------------------------
__global__ __launch_bounds__(256) void scatter_edges(
    const float* __restrict__ trans,  // [4, N, 256]
    const int* __restrict__ srcI,
    const int* __restrict__ dstI,
    const int* __restrict__ etype,
    const float* __restrict__ cnt,    // [N, 4]
    float* __restrict__ u0,           // [N, 256]
    int N, int E) {
    int e = blockIdx.x * 4 + (threadIdx.x >> 6);
    if (e >= E) return;
    int t64 = threadIdx.x & 63;

    int s = srcI[e];
    int d = dstI[e];
    int r = etype[e];
    float c = cnt[d * 4 + r];
    float scale = 1.0f / fmaxf(c, 1.0f);

    const float4* __restrict__ tp =
        (const float4*)(trans + ((size_t)r * N + s) * FEAT);
    float4 v = tp[t64];

    float* __restrict__ op = u0 + (size_t)d * FEAT + t64 * 4;
    atomicAdd(op + 0, v.x * scale);
    atomicAdd(op + 1, v.y * scale);
    atomicAdd(op + 2, v.z * scale);
    atomicAdd(op + 3, v.w * scale);
}

// ---------------------------------------------------------------------------
// Kernel 4: gating epilogue. One wave32 per node.
//   a = sigmoid( dot(u0, attW[0:256]) + dot(x, attW[256:512]) + attB )
//   h = tanh(u0) * a + x * (1 - a)
// ---------------------------------------------------------------------------
__global__ __launch_bounds__(256) void gate_nodes(
    const float* __restrict__ u0,
    const float* __restrict__ X,
    const float* __restrict__ attW,   // [512]
    const float* __restrict__ attB,   // [1]
    float* __restrict__ out,          // [N, 256]
    int N) {
    int node = blockIdx.x * 8 + (threadIdx.x >> 5);
    int lane = threadIdx.x & 31;
    if (node >= N) return;

    const float4* __restrict__ up = (const float4*)(u0 + (size_t)node * FEAT);
    const float4* __restrict__ xp = (const float4*)(X + (size_t)node * FEAT);
    const float4* __restrict__ w0 = (const float4*)(attW);        // u0 part
    const float4* __restrict__ w1 = (const float4*)(attW + FEAT); // x part

    float4 uu[2], xx[2];
    float dot = 0.0f;
#pragma unroll
    for (int i = 0; i < 2; ++i) {
        int idx = lane + 32 * i;
        uu[i] = up[idx];
        xx[i] = xp[idx];
        float4 a = w0[idx];
        float4 b = w1[idx];
        dot += uu[i].x * a.x + uu[i].y * a.y + uu[i].z * a.z + uu[i].w * a.w;
        dot += xx[i].x * b.x + xx[i].y * b.y + xx[i].z * b.z + xx[i].w * b.w;
    }
    // wave32 butterfly reduction
#pragma unroll
    for (int off = 16; off > 0; off >>= 1) dot += __shfl_xor(dot, off, 32);

    float g = 1.0f / (1.0f + __expf(-(dot + attB[0])));
    float omg = 1.0f - g;

    float4* __restrict__ op = (float4*)(out + (size_t)node * FEAT);
#pragma unroll
    for (int i = 0; i < 2; ++i) {
        int idx = lane + 32 * i;
        float4 h;
        h.x = tanhf(uu[i].x) * g + xx[i].x * omg;
        h.y = tanhf(uu[i].y) * g + xx[i].y * omg;
        h.z = tanhf(uu[i].z) * g + xx[i].z * omg;
        h.w = tanhf(uu[i].w) * g + xx[i].w * omg;
        op[idx] = h;
    }
}

// ---------------------------------------------------------------------------
// Launch. Inputs (setup_inputs order):
//  0 node_features [N,256] f32   4 root_weight [256,256] f32
//  1 edge_index    [2,E]  int    5 rgcn_bias   [256]     f32
//  2 edge_type     [E]    int    6 att_weight  [1,512]   f32
//  3 rel_weight [4,256,256] f32  7 att_bias    [1]       f32
// Workspace: trans (4*N*256 f32) | u0 (N*256 f32) | cnt (N*4 f32)  ~= 257 MB.
// ---------------------------------------------------------------------------
extern "C" void kernel_launch(void* const* d_in, const int* in_sizes, int n_in,
                              void* d_out, int out_size, void* d_ws, size_t ws_size,
                              hipStream_t stream) {
    const float* X     = (const float*)d_in[0];
    const int*   eidx  = (const int*)d_in[1];
    const int*   etype = (const int*)d_in[2];
    const float* relW  = (const float*)d_in[3];
    const float* rootW = (const float*)d_in[4];
    const float* bias  = (const float*)d_in[5];
    const float* attW  = (const float*)d_in[6];
    const float* attB  = (const float*)d_in[7];

    const int N = in_sizes[0] / FEAT;
    const int E = in_sizes[1] / 2;
    const int* srcI = eidx;       // edge_index[0, :]
    const int* dstI = eidx + E;   // edge_index[1, :]

    char* ws = (char*)d_ws;
    float* trans = (float*)ws;                                           // 4*N*256
    float* u0    = (float*)(ws + (size_t)4 * N * FEAT * sizeof(float));  // N*256
    float* cnt   = (float*)(ws + (size_t)5 * N * FEAT * sizeof(float));  // N*4

    // 1) zero degree counts, then count per-(dst, relation) in-degrees
    int nz = N * 4;
    zero_f32<<<(nz + 255) / 256, 256, 0, stream>>>(cnt, nz);
    count_edges<<<(E + 255) / 256, 256, 0, stream>>>(dstI, etype, cnt, E);

    // 2) five fused GEMMs via fp32 WMMA (u0 = X@rootW+b, trans[r] = X@relW[r])
    dim3 gg((N + 15) / 16, 5);
    rgcn_gemm_wmma<<<gg, 128, 0, stream>>>(X, rootW, relW, bias, u0, trans, N);

    // 3) edge scatter: u0[dst] += trans[r][src] / max(cnt[dst][r], 1)
    scatter_edges<<<(E + 3) / 4, 256, 0, stream>>>(trans, srcI, dstI, etype,
                                                   cnt, u0, N, E);

    // 4) attention-gated blend epilogue -> d_out
    gate_nodes<<<(N + 7) / 8, 256, 0, stream>>>(u0, X, attW, attB,
                                                (float*)d_out, N);
}